// ClassifierCaps_57114475102339
// MI455X (gfx1250) — compile-verified
//
#include <hip/hip_runtime.h>
#include <math.h>

// ---------------- problem constants (from reference) ----------------
#define B_    64
#define N_    1152
#define DIN   8
#define C_    128
#define DOUT  16
#define CH    16                 // n-chunk staged in LDS (16 * 128 floats = 8KB)
#define NCHUNK (N_ / CH)         // 72
#define LEAKY 0.01f

typedef float v2f __attribute__((ext_vector_type(2)));
typedef float v8f __attribute__((ext_vector_type(8)));

// D = A(16x4 f32) * B(4x16 f32) + C(16x16 f32), wave32, full fp32.
__device__ __forceinline__ v8f wmma_16x16x4_f32(v2f a, v2f b, v8f c) {
#if defined(__HIP_DEVICE_COMPILE__)
  // 8 args: (neg_a, A, neg_b, B, c_mod, C, reuse_a, reuse_b)
  return __builtin_amdgcn_wmma_f32_16x16x4_f32(false, a, false, b, (short)0, c,
                                               false, false);
#else
  (void)a; (void)b;
  return c;
#endif
}

// one xor-butterfly step within a 16-lane row via DPP ROW_XMASK (pure VALU)
template <int MASK>
__device__ __forceinline__ float row_xor_add(float t) {
#if defined(__HIP_DEVICE_COMPILE__)
  int p = __builtin_amdgcn_update_dpp(0, __float_as_int(t), 0x160 | MASK,
                                      0xf, 0xf, true);
  return t + __int_as_float(p);
#else
  return t;
#endif
}

// sum across the 16 lanes of a half-wave group; result replicated in group
__device__ __forceinline__ float rsum16(float t) {
  t = row_xor_add<1>(t);
  t = row_xor_add<2>(t);
  t = row_xor_add<4>(t);
  t = row_xor_add<8>(t);
  return t;
}

// CDNA5 async global->LDS copy (16B per lane), tracked by ASYNCcnt.
// No data VGPRs involved -- this is the gfx1250 LDS-DMA path.
__device__ __forceinline__ void async_copy_b128(unsigned lds_off,
                                                const float* gsrc) {
#if defined(__HIP_DEVICE_COMPILE__)
  asm volatile("global_load_async_to_lds_b128 %0, %1, off"
               :
               : "v"(lds_off), "v"((unsigned long long)(uintptr_t)gsrc)
               : "memory");
#else
  (void)lds_off; (void)gsrc;
#endif
}
__device__ __forceinline__ void wait_async() {
#if defined(__HIP_DEVICE_COMPILE__)
  asm volatile("s_wait_asynccnt 0x0" ::: "memory");
#endif
}

// ---------------- kernel 0: zero the per-batch sums ----------------
__global__ void zero_sums(float* __restrict__ sums) {
  if (threadIdx.x < B_) sums[threadIdx.x] = 0.f;
}

// ---------------- kernel 1: squash(x) and transpose to n-major ------
// x: [B][N][8]  ->  xt: [N][B][8]   (A-operand friendly layout)
__global__ void squash_xt(const float* __restrict__ x, float* __restrict__ xt) {
  int t = blockIdx.x * blockDim.x + threadIdx.x;
  if (t >= B_ * N_) return;
  int b = t / N_, n = t % N_;
  const float* p = x + (size_t)t * DIN;
  float v[DIN];
  float sq = 0.f;
#pragma unroll
  for (int i = 0; i < DIN; ++i) { v[i] = p[i]; sq += v[i] * v[i]; }
  float scale = sq / ((1.f + sq) * sqrtf(sq));
  float* q = xt + ((size_t)n * B_ + b) * DIN;
#pragma unroll
  for (int i = 0; i < DIN; ++i) q[i] = v[i] * scale;
}

// ---------------- kernel 2: fused u_hat + 3 routing iterations ------
// One block per class c (W[c] slice stays L2/LDS resident across 3 sweeps).
// Wave w handles batch rows [16w, 16w+16). u_hat is never written to HBM.
// W chunks are double-buffered in LDS via async LDS-DMA (ASYNCcnt).
__global__ __launch_bounds__(128, 1) void caps_route(
    const float* __restrict__ W,    // [C][N][8][16]
    const float* __restrict__ xt,   // [N][64][8]  (squashed)
    float* __restrict__ out,        // [B][C][16]  (unnormalized after this)
    float* __restrict__ sums)       // [B]
{
  __shared__ float lB[2][CH * DIN * DOUT];   // 2 x 8 KB, shared by 4 waves

  const int c    = blockIdx.x;
  const int tid  = threadIdx.x;
  const int lane = tid & 31;
  const int wave = tid >> 5;
  const int lo   = lane & 15;     // N-index (dout) in C/D layout
  const int hi   = lane >> 4;     // half-wave select (rows +8, K +2)
  const int b0   = wave * 16;

  const float* Wc = W + (size_t)c * N_ * (DIN * DOUT);
  const unsigned ldsbase = (unsigned)(uintptr_t)(&lB[0][0]);

  // issue async copy of one 8KB W chunk into LDS buffer `buf`
  auto stage_async = [&](int ch, int buf) {
    const float* src = Wc + (size_t)ch * CH * DIN * DOUT + tid * 4;
    unsigned l = ldsbase + (unsigned)buf * (CH * DIN * DOUT * 4) + tid * 16;
#pragma unroll
    for (int r = 0; r < 4; ++r)
      async_copy_b128(l + r * 2048, src + r * 512);
  };

  // A operand (16x4 f32): lanes0-15 K={0,1}, lanes16-31 K={2,3}
  // B operand (4x16 f32): VGPR0 K={0|2}, VGPR1 K={1|3}
#define LOAD_OPERANDS(lbuf, n, nn)                                          \
  const float* ap = xt + ((size_t)(n)*B_ + b0 + lo) * DIN + hi * 2;         \
  v2f a0; a0.x = ap[0]; a0.y = ap[1];                                       \
  v2f a1; a1.x = ap[4]; a1.y = ap[5];                                       \
  const float* bp = (lbuf) + (nn) * (DIN * DOUT) + (hi * 2) * DOUT + lo;    \
  v2f bk0; bk0.x = bp[0];        bk0.y = bp[DOUT];                          \
  v2f bk1; bk1.x = bp[4 * DOUT]; bk1.y = bp[5 * DOUT];

  // V[j]: accumulated v in C/D layout (row = b0+j+8*hi, col = lo);
  // after the last iteration it holds the final v.
  float V[8];

  // ---------------- iteration 0: uniform coefficients -----------------
  // s0 = (1/N) Sigma_n u_hat : two independent WMMA accumulation chains
  {
    v8f acc0 = {}, acc1 = {};
    stage_async(0, 0);
    wait_async();
    __syncthreads();
    for (int ch = 0; ch < NCHUNK; ++ch) {
      if (ch + 1 < NCHUNK) stage_async(ch + 1, (ch + 1) & 1);
      const float* lbuf = lB[ch & 1];
#pragma unroll 4
      for (int nn = 0; nn < CH; ++nn) {
        LOAD_OPERANDS(lbuf, ch * CH + nn, nn)
        acc0 = wmma_16x16x4_f32(a0, bk0, acc0);
        acc1 = wmma_16x16x4_f32(a1, bk1, acc1);
      }
      if (ch + 1 < NCHUNK) { wait_async(); __syncthreads(); }
    }
#pragma unroll
    for (int j = 0; j < 8; ++j) {
      float s  = (acc0[j] + acc1[j]) * (1.0f / N_);
      float sq = rsum16(s * s);
      float sc = sq / ((1.f + sq) * sqrtf(sq));
      V[j] = s * sc;                       // V = v0
    }
  }

  // ---------------- iterations 1..2: softmax routing -------------------
  for (int iter = 1; iter < 3; ++iter) {
    float sa[8], Zr[8];
#pragma unroll
    for (int j = 0; j < 8; ++j) { sa[j] = 0.f; Zr[j] = 0.f; }

    stage_async(0, 0);
    wait_async();
    __syncthreads();
    for (int ch = 0; ch < NCHUNK; ++ch) {
      if (ch + 1 < NCHUNK) stage_async(ch + 1, (ch + 1) & 1);
      const float* lbuf = lB[ch & 1];
#pragma unroll 4
      for (int nn = 0; nn < CH; ++nn) {
        LOAD_OPERANDS(lbuf, ch * CH + nn, nn)
        v8f z = {};
        v8f u = wmma_16x16x4_f32(a0, bk0, z);
        u     = wmma_16x16x4_f32(a1, bk1, u);
#pragma unroll
        for (int j = 0; j < 8; ++j) {
          // routing logit b_n = u_hat[row] . V[row]   (shift-free softmax:
          // operands are squash-normalized so logits are bounded)
          float d = rsum16(u[j] * V[j]);
          float w = __expf(d);
          Zr[j] += w;
          sa[j] = fmaf(w, u[j], sa[j]);
        }
      }
      if (ch + 1 < NCHUNK) { wait_async(); __syncthreads(); }
    }

    // s -> v = squash(s);  V accumulates (logit identity b = u_hat . Sigma v),
    // and on the final iteration V is replaced by the result v.
#pragma unroll
    for (int j = 0; j < 8; ++j) {
      float s  = sa[j] / Zr[j];
      float sq = rsum16(s * s);
      float sc = sq / ((1.f + sq) * sqrtf(sq));
      float v  = s * sc;
      V[j] = (iter == 1) ? (V[j] + v) : v;
    }
    __syncthreads();   // protect lB[0] before next iteration restages it
  }

  // finalize: leaky_relu^2, store unnormalized, accumulate per-batch sum
#pragma unroll
  for (int j = 0; j < 8; ++j) {
    float v = V[j];
    float f = (v < 0.f) ? (LEAKY * v) : v;
    f = f * f;
    int b = b0 + j + 8 * hi;
    out[((size_t)b * C_ + c) * DOUT + lo] = f;
    float rs = rsum16(f);
    if (lo == 0) atomicAdd(&sums[b], rs);
  }
}

// ---------------- kernel 3: divide by per-batch sum ------------------
__global__ void normalize_out(float* __restrict__ out,
                              const float* __restrict__ sums) {
  int idx = blockIdx.x * blockDim.x + threadIdx.x;
  if (idx >= B_ * C_ * DOUT) return;
  int b = idx / (C_ * DOUT);
  out[idx] /= sums[b];
}

// ---------------- launcher -------------------------------------------
extern "C" void kernel_launch(void* const* d_in, const int* in_sizes, int n_in,
                              void* d_out, int out_size, void* d_ws,
                              size_t ws_size, hipStream_t stream) {
  (void)in_sizes; (void)n_in; (void)out_size; (void)ws_size;
  const float* x = (const float*)d_in[0];   // [64,1152,8]
  const float* W = (const float*)d_in[1];   // [128,1152,8,16]
  float* out = (float*)d_out;               // [64,128,16]

  float* xt   = (float*)d_ws;                      // N*B*8 floats (2.36 MB)
  float* sums = xt + (size_t)N_ * B_ * DIN;        // 64 floats

  zero_sums<<<1, 64, 0, stream>>>(sums);
  squash_xt<<<(B_ * N_ + 255) / 256, 256, 0, stream>>>(x, xt);
  caps_route<<<C_, 128, 0, stream>>>(W, xt, out, sums);
  normalize_out<<<(B_ * C_ * DOUT + 255) / 256, 256, 0, stream>>>(out, sums);
}